// FaultGNN_44590350467555
// MI455X (gfx1250) — compile-verified
//
#include <hip/hip_runtime.h>

#define HID 128
#define LN_EPS 1e-5f

typedef __attribute__((ext_vector_type(2))) float v2f;
typedef __attribute__((ext_vector_type(8))) float v8f;

// ---------------------------------------------------------------- utilities

__global__ void zero_kernel(float* __restrict__ p, int n) {
    int i = blockIdx.x * blockDim.x + threadIdx.x;
    int stride = gridDim.x * blockDim.x;
    for (; i < n; i += stride) p[i] = 0.0f;
}

__global__ void degree_kernel(const int* __restrict__ dst, float* __restrict__ deg, int E) {
    int i = blockIdx.x * blockDim.x + threadIdx.x;
    if (i < E) atomicAdd(deg + dst[i], 1.0f);
}

// one wave per edge: gather 128 floats of h[src], atomically add into agg[dst]
__global__ void scatter_kernel(const float* __restrict__ h, const int* __restrict__ src,
                               const int* __restrict__ dst, float* __restrict__ agg, int E) {
    int wave = (blockIdx.x * blockDim.x + threadIdx.x) >> 5;
    int lane = threadIdx.x & 31;
    if (wave >= E) return;
    int s = src[wave];
    int d = dst[wave];
    const float4 v = *(const float4*)(h + (size_t)s * HID + lane * 4);
    float* b = agg + (size_t)d * HID + lane * 4;
    atomicAdd(b + 0, v.x);
    atomicAdd(b + 1, v.y);
    atomicAdd(b + 2, v.z);
    atomicAdd(b + 3, v.w);
}

// agg[n][:] /= max(deg[n], 1); one thread per float4 (32 threads per node row)
__global__ void scale_kernel(float* __restrict__ agg, const float* __restrict__ deg, int N) {
    int i = blockIdx.x * blockDim.x + threadIdx.x;
    int total = N * (HID / 4);
    if (i >= total) return;
    int node = i >> 5;
    float r = 1.0f / fmaxf(deg[node], 1.0f);
    float4* p = (float4*)agg + i;
    float4 v = *p;
    v.x *= r; v.y *= r; v.z *= r; v.w *= r;
    *p = v;
}

// LayerNorm + LeakyReLU(0.2), in place; one wave per 128-wide row
__global__ void ln_leaky_kernel(float* __restrict__ h, const float* __restrict__ g,
                                const float* __restrict__ b, int N) {
    int wave = (blockIdx.x * blockDim.x + threadIdx.x) >> 5;
    int lane = threadIdx.x & 31;
    if (wave >= N) return;
    float* row = h + (size_t)wave * HID;
    float4 v = *(const float4*)(row + lane * 4);
    float s  = v.x + v.y + v.z + v.w;
    float sq = v.x * v.x + v.y * v.y + v.z * v.z + v.w * v.w;
#pragma unroll
    for (int off = 16; off > 0; off >>= 1) {
        s  += __shfl_xor(s,  off, 32);
        sq += __shfl_xor(sq, off, 32);
    }
    float mu  = s * (1.0f / HID);
    float var = sq * (1.0f / HID) - mu * mu;
    float rs  = rsqrtf(var + LN_EPS);
    float4 gg = *(const float4*)(g + lane * 4);
    float4 bb = *(const float4*)(b + lane * 4);
    float4 o;
    o.x = (v.x - mu) * rs * gg.x + bb.x;
    o.y = (v.y - mu) * rs * gg.y + bb.y;
    o.z = (v.z - mu) * rs * gg.z + bb.z;
    o.w = (v.w - mu) * rs * gg.w + bb.w;
    o.x = o.x > 0.0f ? o.x : 0.2f * o.x;
    o.y = o.y > 0.0f ? o.y : 0.2f * o.y;
    o.z = o.z > 0.0f ? o.z : 0.2f * o.z;
    o.w = o.w > 0.0f ? o.w : 0.2f * o.w;
    *(float4*)(row + lane * 4) = o;
}

// ---------------------------------------------------------------- WMMA GEMM
// out[M x 128] = A1[M x 128] @ W1[128 x 128] (+ A2 @ W2) + bias  (+BN+ReLU)
// One wave computes a 16-row x 128-col strip via 8 accumulator tiles of
// V_WMMA_F32_16X16X4_F32 (full fp32 precision, K stepped by 4).
//
// A 16x4 f32 layout: lanes 0-15 -> M=lane, VGPR0=K0,VGPR1=K1;
//                    lanes 16-31 -> M=lane-16, VGPR0=K2,VGPR1=K3.
// B 4x16 mirrored: lane -> N, VGPR pair -> K-half.
// C 16x16: lane -> N (mod 16), VGPR r -> M = r (lanes<16) / r+8 (lanes>=16).
template <bool DUAL, bool BNRELU>
__global__ void sage_gemm_kernel(const float* __restrict__ A1, const float* __restrict__ W1,
                                 const float* __restrict__ A2, const float* __restrict__ W2,
                                 const float* __restrict__ bias,
                                 const float* __restrict__ bn_g, const float* __restrict__ bn_b,
                                 const float* __restrict__ bn_m, const float* __restrict__ bn_v,
                                 float* __restrict__ out, int M) {
    int wave = (blockIdx.x * blockDim.x + threadIdx.x) >> 5;
    int lane = threadIdx.x & 31;
    if (wave * 16 >= M) return;  // wave-uniform: EXEC stays all-ones for WMMA

    const int row16 = lane & 15;
    const int khalf = (lane >> 4) * 2;

    v8f acc[8];
#pragma unroll
    for (int j = 0; j < 8; j++) acc[j] = (v8f)(0.0f);

    const float* a1row = A1 + ((size_t)wave * 16 + row16) * HID + khalf;
#pragma unroll 4
    for (int k0 = 0; k0 < HID; k0 += 4) {
        v2f a = *(const v2f*)(a1row + k0);
#pragma unroll
        for (int j = 0; j < 8; j++) {
            int n = j * 16 + row16;
            v2f b;
            b.x = W1[(size_t)(k0 + khalf) * HID + n];
            b.y = W1[(size_t)(k0 + khalf + 1) * HID + n];
            acc[j] = __builtin_amdgcn_wmma_f32_16x16x4_f32(
                false, a, false, b, (short)0, acc[j], false, false);
        }
    }
    if (DUAL) {
        const float* a2row = A2 + ((size_t)wave * 16 + row16) * HID + khalf;
#pragma unroll 4
        for (int k0 = 0; k0 < HID; k0 += 4) {
            v2f a = *(const v2f*)(a2row + k0);
#pragma unroll
            for (int j = 0; j < 8; j++) {
                int n = j * 16 + row16;
                v2f b;
                b.x = W2[(size_t)(k0 + khalf) * HID + n];
                b.y = W2[(size_t)(k0 + khalf + 1) * HID + n];
                acc[j] = __builtin_amdgcn_wmma_f32_16x16x4_f32(
                    false, a, false, b, (short)0, acc[j], false, false);
            }
        }
    }

    // epilogue: each lane owns one column per tile, 8 rows per VGPR index
    int mbase = wave * 16 + ((lane < 16) ? 0 : 8);
#pragma unroll
    for (int j = 0; j < 8; j++) {
        int n = j * 16 + row16;
        float bi = bias[n];
        float bn_a = 1.0f, bn_c = 0.0f;
        if (BNRELU) {
            bn_a = bn_g[n] * rsqrtf(bn_v[n] + LN_EPS);
            bn_c = bn_b[n] - bn_m[n] * bn_a;
        }
        float* col = out + (size_t)mbase * HID + n;
#pragma unroll
        for (int r = 0; r < 8; r++) {
            float val = acc[j][r] + bi;
            if (BNRELU) val = fmaxf(bn_a * val + bn_c, 0.0f);
            col[(size_t)r * HID] = val;
        }
    }
}

// ---------------------------------------------------------------- final layer
// out[n][0:2] = agg[n] @ Wl2 + bl2 + h[n] @ Wr2 ; one wave per node
__global__ void final_kernel(const float* __restrict__ agg, const float* __restrict__ h,
                             const float* __restrict__ Wl, const float* __restrict__ Wr,
                             const float* __restrict__ bl, float* __restrict__ out, int N) {
    int wave = (blockIdx.x * blockDim.x + threadIdx.x) >> 5;
    int lane = threadIdx.x & 31;
    if (wave >= N) return;
    const float4 a = *(const float4*)(agg + (size_t)wave * HID + lane * 4);
    const float4 x = *(const float4*)(h   + (size_t)wave * HID + lane * 4);
    // W is [128 x 2] row-major: rows k..k+3 for this lane = 8 consecutive floats
    const float4 wl01 = *(const float4*)(Wl + lane * 8);
    const float4 wl23 = *(const float4*)(Wl + lane * 8 + 4);
    const float4 wr01 = *(const float4*)(Wr + lane * 8);
    const float4 wr23 = *(const float4*)(Wr + lane * 8 + 4);
    float s0 = a.x * wl01.x + a.y * wl01.z + a.z * wl23.x + a.w * wl23.z
             + x.x * wr01.x + x.y * wr01.z + x.z * wr23.x + x.w * wr23.z;
    float s1 = a.x * wl01.y + a.y * wl01.w + a.z * wl23.y + a.w * wl23.w
             + x.x * wr01.y + x.y * wr01.w + x.z * wr23.y + x.w * wr23.w;
#pragma unroll
    for (int off = 16; off > 0; off >>= 1) {
        s0 += __shfl_xor(s0, off, 32);
        s1 += __shfl_xor(s1, off, 32);
    }
    if (lane == 0) {
        out[(size_t)wave * 2 + 0] = s0 + bl[0];
        out[(size_t)wave * 2 + 1] = s1 + bl[1];
    }
}

// ---------------------------------------------------------------- launch

extern "C" void kernel_launch(void* const* d_in, const int* in_sizes, int n_in,
                              void* d_out, int out_size, void* d_ws, size_t ws_size,
                              hipStream_t stream) {
    const float* x     = (const float*)d_in[0];
    const int*   ei    = (const int*)d_in[1];
    const float* ft_W  = (const float*)d_in[2];
    const float* ft_b  = (const float*)d_in[3];
    const float* ln_g  = (const float*)d_in[4];
    const float* ln_b  = (const float*)d_in[5];
    const float* Wl0   = (const float*)d_in[6];
    const float* Wr0   = (const float*)d_in[7];
    const float* bl0   = (const float*)d_in[8];
    const float* bn0_g = (const float*)d_in[9];
    const float* bn0_b = (const float*)d_in[10];
    const float* bn0_m = (const float*)d_in[11];
    const float* bn0_v = (const float*)d_in[12];
    const float* Wl1   = (const float*)d_in[13];
    const float* Wr1   = (const float*)d_in[14];
    const float* bl1   = (const float*)d_in[15];
    const float* bn1_g = (const float*)d_in[16];
    const float* bn1_b = (const float*)d_in[17];
    const float* bn1_m = (const float*)d_in[18];
    const float* bn1_v = (const float*)d_in[19];
    const float* Wl2   = (const float*)d_in[20];
    const float* Wr2   = (const float*)d_in[21];
    const float* bl2   = (const float*)d_in[22];
    float* out = (float*)d_out;

    const int N = in_sizes[0] / HID;   // 50000
    const int E = in_sizes[1] / 2;     // 800000
    const int* src = ei;
    const int* dst = ei + E;

    // workspace layout
    char* ws = (char*)d_ws;
    size_t degBytes = (((size_t)N * 4) + 255) / 256 * 256;
    float* deg = (float*)ws;
    float* agg = (float*)(ws + degBytes);
    float* hA  = agg + (size_t)N * HID;
    float* hB  = hA + (size_t)N * HID;

    const int TPB = 256;
    const int NH = N * HID;
    int mtiles = (N + 15) / 16;
    int gemmBlocks = (mtiles + 3) / 4;                 // 4 waves / block
    int rowBlocks = (N + 3) / 4;                       // wave-per-row kernels
    int scatBlocks = ((size_t)E * 32 + TPB - 1) / TPB; // wave-per-edge
    int scaleBlocks = (N * 32 + TPB - 1) / TPB;

    // degrees
    zero_kernel<<<256, TPB, 0, stream>>>(deg, N);
    degree_kernel<<<(E + TPB - 1) / TPB, TPB, 0, stream>>>(dst, deg, E);

    // feature transform: hA = LeakyReLU(LN(x @ ft_W + ft_b))
    sage_gemm_kernel<false, false><<<gemmBlocks, 128, 0, stream>>>(
        x, ft_W, nullptr, nullptr, ft_b, nullptr, nullptr, nullptr, nullptr, hA, N);
    ln_leaky_kernel<<<rowBlocks, 128, 0, stream>>>(hA, ln_g, ln_b, N);

    // SAGE layer 0: hB = ReLU(BN0(mean_agg(hA) @ Wl0 + bl0 + hA @ Wr0))
    zero_kernel<<<2048, TPB, 0, stream>>>(agg, NH);
    scatter_kernel<<<scatBlocks, TPB, 0, stream>>>(hA, src, dst, agg, E);
    scale_kernel<<<scaleBlocks, TPB, 0, stream>>>(agg, deg, N);
    sage_gemm_kernel<true, true><<<gemmBlocks, 128, 0, stream>>>(
        agg, Wl0, hA, Wr0, bl0, bn0_g, bn0_b, bn0_m, bn0_v, hB, N);

    // SAGE layer 1: hA = ReLU(BN1(mean_agg(hB) @ Wl1 + bl1 + hB @ Wr1))
    zero_kernel<<<2048, TPB, 0, stream>>>(agg, NH);
    scatter_kernel<<<scatBlocks, TPB, 0, stream>>>(hB, src, dst, agg, E);
    scale_kernel<<<scaleBlocks, TPB, 0, stream>>>(agg, deg, N);
    sage_gemm_kernel<true, true><<<gemmBlocks, 128, 0, stream>>>(
        agg, Wl1, hB, Wr1, bl1, bn1_g, bn1_b, bn1_m, bn1_v, hA, N);

    // final SAGE layer -> logits
    zero_kernel<<<2048, TPB, 0, stream>>>(agg, NH);
    scatter_kernel<<<scatBlocks, TPB, 0, stream>>>(hA, src, dst, agg, E);
    scale_kernel<<<scaleBlocks, TPB, 0, stream>>>(agg, deg, N);
    final_kernel<<<rowBlocks, 128, 0, stream>>>(agg, hA, Wl2, Wr2, bl2, out, N);
}